// InverseHadamardTransform_53120155517139
// MI455X (gfx1250) — compile-verified
//
#include <hip/hip_runtime.h>
#include <hip/hip_bf16.h>

// FWHT-4096 on MI455X (gfx1250): H4096 = H16 (x) H16 (x) H16, each 16-axis
// contraction done with V_WMMA_F32_16X16X4_F32 (4 K-chunks of K=4).
// Memory-bound: 512 MB total traffic -> ~22us floor at 23.3 TB/s.
// Global<->LDS staging uses gfx1250 async-to-LDS (ASYNCcnt) b128 ops when
// the toolchain exposes them, else b128 VGPR staging.

typedef float v2f __attribute__((ext_vector_type(2)));
typedef float v4f __attribute__((ext_vector_type(4)));
typedef float v8f __attribute__((ext_vector_type(8)));
typedef int   v4i __attribute__((ext_vector_type(4)));

typedef __attribute__((address_space(1))) v4i* gptr_v4i;  // global int4*
typedef __attribute__((address_space(3))) v4i* lptr_v4i;  // LDS int4*

#define FWHT_N  4096
#define LDS_SA  340   // a-slice stride: 0 mod 4 (b128 align), breaks bank reuse across a-halves
#define LDS_SB  20    // b-row stride: 16 + 4 pad, 0 mod 4 (b128 align)

#if __has_builtin(__builtin_amdgcn_global_load_async_to_lds_b128) && \
    __has_builtin(__builtin_amdgcn_global_store_async_from_lds_b128) && \
    __has_builtin(__builtin_amdgcn_s_wait_asynccnt)
#define USE_ASYNC_LDS 1
#else
#define USE_ASYNC_LDS 0
#endif

__global__ __launch_bounds__(256)
void fwht4096_wmma_kernel(const float* __restrict__ in,
                          float* __restrict__ out) {
    __shared__ float s[16 * LDS_SA];   // 21760 bytes

    const int row  = blockIdx.x;
    const float* __restrict__ src = in  + (size_t)row * FWHT_N;
    float* __restrict__       dst = out + (size_t)row * FWHT_N;

    const int tid  = threadIdx.x;       // 0..255
    const int w    = tid >> 5;          // wave id 0..7
    const int lane = tid & 31;
    const int half = lane >> 4;         // 0: lanes 0-15, 1: lanes 16-31
    const int l15  = lane & 15;

    // float4-granularity staging map: f = q*256 + tid, element i = 4*f
    //   a = q*4 + (tid>>6), b = (tid>>2)&15, c = (tid&3)*4
    const int st_a0 = tid >> 6;
    const int st_bc = ((tid >> 2) & 15) * LDS_SB + (tid & 3) * 4;

    // ---- stage-in: global -> LDS (b128 per lane) ----
#pragma unroll
    for (int q = 0; q < 4; ++q) {
        const int lidx = (q * 4 + st_a0) * LDS_SA + st_bc;
        const int f    = q * 256 + tid;
#if USE_ASYNC_LDS
        __builtin_amdgcn_global_load_async_to_lds_b128(
            (gptr_v4i)(src + (size_t)f * 4),
            (lptr_v4i)&s[lidx], 0, 0);
#else
        const v4f d = __builtin_nontemporal_load((const v4f*)src + f);
        s[lidx + 0] = d[0];
        s[lidx + 1] = d[1];
        s[lidx + 2] = d[2];
        s[lidx + 3] = d[3];
#endif
    }

    // ---- A operand: H16 chunks, per documented 16x4 f32 A layout ----
    // lane 0-15 : M = lane,    VGPR v -> K = 4j + v
    // lane 16-31: M = lane-16, VGPR v -> K = 4j + 2 + v
    // H16[m,k] = (-1)^popc(m & k)
    v2f A[4];
#pragma unroll
    for (int j = 0; j < 4; ++j) {
#pragma unroll
        for (int v = 0; v < 2; ++v) {
            const int K = 4 * j + 2 * half + v;
            A[j][v] = (__popc(l15 & K) & 1) ? -1.0f : 1.0f;
        }
    }

#if USE_ASYNC_LDS
    __builtin_amdgcn_s_wait_asynccnt(0);   // own async loads landed in LDS
#endif
    __syncthreads();

    // ---- stage 1: contract c.  D[c',b] = sum_c H[c',c] * X[a, b(=N), c(=K)] ----
#pragma unroll
    for (int tt = 0; tt < 2; ++tt) {
        const int t = w + 8 * tt;           // tile = a
        v8f acc = {};
#pragma unroll
        for (int j = 0; j < 4; ++j) {
            const int k0 = 4 * j + 2 * half;
            v2f b;
            b[0] = s[t * LDS_SA + l15 * LDS_SB + k0 + 0];
            b[1] = s[t * LDS_SA + l15 * LDS_SB + k0 + 1];
            acc = __builtin_amdgcn_wmma_f32_16x16x4_f32(
                false, A[j], false, b, (short)0, acc, false, false);
        }
#pragma unroll
        for (int v = 0; v < 8; ++v)        // D: M = v + 8*half (=c'), N = l15 (=b)
            s[t * LDS_SA + l15 * LDS_SB + (v + 8 * half)] = acc[v];
    }

    __syncthreads();

    // ---- stage 2: contract b.  D[b',c'] = sum_b H[b',b] * Y1[a, b(=K), c'(=N)] ----
#pragma unroll
    for (int tt = 0; tt < 2; ++tt) {
        const int t = w + 8 * tt;           // tile = a
        v8f acc = {};
#pragma unroll
        for (int j = 0; j < 4; ++j) {
            const int k0 = 4 * j + 2 * half;
            v2f b;
            b[0] = s[t * LDS_SA + (k0 + 0) * LDS_SB + l15];
            b[1] = s[t * LDS_SA + (k0 + 1) * LDS_SB + l15];
            acc = __builtin_amdgcn_wmma_f32_16x16x4_f32(
                false, A[j], false, b, (short)0, acc, false, false);
        }
#pragma unroll
        for (int v = 0; v < 8; ++v)        // D: M = b' = v + 8*half, N = c' = l15
            s[t * LDS_SA + (v + 8 * half) * LDS_SB + l15] = acc[v];
    }

    __syncthreads();

    // ---- stage 3: contract a.  D[a',c'] = sum_a H[a',a] * Y2[a(=K), b', c'(=N)] ----
    const float scale = 0.015625f;          // 1/sqrt(4096)
#pragma unroll
    for (int tt = 0; tt < 2; ++tt) {
        const int t = w + 8 * tt;           // tile = b'
        v8f acc = {};
#pragma unroll
        for (int j = 0; j < 4; ++j) {
            const int k0 = 4 * j + 2 * half;
            v2f b;
            b[0] = s[(k0 + 0) * LDS_SA + t * LDS_SB + l15];
            b[1] = s[(k0 + 1) * LDS_SA + t * LDS_SB + l15];
            acc = __builtin_amdgcn_wmma_f32_16x16x4_f32(
                false, A[j], false, b, (short)0, acc, false, false);
        }
#pragma unroll
        for (int v = 0; v < 8; ++v)        // D: M = a' = v + 8*half, N = c' = l15
            s[(v + 8 * half) * LDS_SA + t * LDS_SB + l15] = acc[v] * scale;
    }

    __syncthreads();

    // ---- stage-out: LDS -> global (b128 per lane) ----
#pragma unroll
    for (int q = 0; q < 4; ++q) {
        const int lidx = (q * 4 + st_a0) * LDS_SA + st_bc;
        const int f    = q * 256 + tid;
#if USE_ASYNC_LDS
        __builtin_amdgcn_global_store_async_from_lds_b128(
            (gptr_v4i)(dst + (size_t)f * 4),
            (lptr_v4i)&s[lidx], 0, 0);
#else
        v4f d;
        d[0] = s[lidx + 0];
        d[1] = s[lidx + 1];
        d[2] = s[lidx + 2];
        d[3] = s[lidx + 3];
        __builtin_nontemporal_store(d, (v4f*)dst + f);
#endif
    }
#if USE_ASYNC_LDS
    __builtin_amdgcn_s_wait_asynccnt(0);   // drain before wave retires
#endif
}

extern "C" void kernel_launch(void* const* d_in, const int* in_sizes, int n_in,
                              void* d_out, int out_size, void* d_ws, size_t ws_size,
                              hipStream_t stream) {
    (void)n_in; (void)d_ws; (void)ws_size; (void)out_size;
    const float* x = (const float*)d_in[0];
    float* y = (float*)d_out;
    const int rows = in_sizes[0] / FWHT_N;   // 2*8192 = 16384 rows of 4096
    fwht4096_wmma_kernel<<<dim3(rows), dim3(256), 0, stream>>>(x, y);
}